// Transformer_927712936298
// MI455X (gfx1250) — compile-verified
//
#include <hip/hip_runtime.h>
#include <hip/hip_bf16.h>
#include <math.h>

// ---------------------------------------------------------------------------
// CDNA5 / gfx1250 transformer micro-model.
// N = 4096 tokens, D = 8, 8 layers.
// Flash attention where BOTH products use V_WMMA_F32_16X16X4_F32:
//   S^T  = K_tile(16x8)  x Q^T(8x16)   (2 WMMAs, K=4 each)
//   O^T += V^T(8x16keys) x P(16keysx16q) (4 WMMAs, K=4 each, C-accumulate)
// P is moved from C/D layout to B-operand layout with one shfl_xor(16) +
// one cndmask per register.  wave32, full-EXEC waves as WMMA requires.
// ---------------------------------------------------------------------------

typedef __attribute__((ext_vector_type(2))) float v2f;
typedef __attribute__((ext_vector_type(8))) float v8f;

#define NTOK 4096
#define DDIM 8

// ---- input [8,64,64] -> emb [4096,8] (row major) --------------------------
__global__ __launch_bounds__(256) void k_embed(const float* __restrict__ in,
                                               float* __restrict__ emb) {
  int i = blockIdx.x * 256 + threadIdx.x;     // 0 .. 32767
  int n = i >> 3;
  int d = i & 7;
  emb[i] = in[d * NTOK + n];
}

// ---- per-feature mean / unbiased var over 4096 rows -----------------------
// 256 threads = 8 waves; wave w reduces feature d = w with shfl_xor.
__global__ __launch_bounds__(256) void k_stats(const float* __restrict__ emb,
                                               float* __restrict__ stats) {
  const int t = threadIdx.x;
  const int d = t >> 5;        // uniform within a wave32
  const int g = t & 31;
  float s = 0.f, s2 = 0.f;
  for (int n = g; n < NTOK; n += 32) {
    float x = emb[n * DDIM + d];
    s += x;
    s2 += x * x;
  }
#pragma unroll
  for (int off = 16; off > 0; off >>= 1) {
    s  += __shfl_xor(s, off, 32);
    s2 += __shfl_xor(s2, off, 32);
  }
  if (g == 0) {
    float mean = s * (1.0f / 4096.0f);
    float var  = (s2 - 4096.0f * mean * mean) * (1.0f / 4095.0f);
    stats[d]     = mean;   // mean
    stats[8 + d] = var;    // unbiased var
  }
}

__device__ __forceinline__ float nan_to_num(float r) {
  if (r != r) return 0.0f;                       // NaN -> 0
  if (r >  3.4028235e38f) return  3.4028235e38f; // +inf -> max
  if (r < -3.4028235e38f) return -3.4028235e38f; // -inf -> min
  return r;
}

// ---- norm + Q,K,V projections (1/sqrt(8) folded into Q) -------------------
__global__ __launch_bounds__(256) void k_qkv(
    const float* __restrict__ emb, const float* __restrict__ stats,
    const float* __restrict__ qw, const float* __restrict__ kw,
    const float* __restrict__ vw, float* __restrict__ q,
    float* __restrict__ k, float* __restrict__ v) {
  const int n = blockIdx.x * 256 + threadIdx.x;
  if (n >= NTOK) return;
  float e[8], nrm[8];
#pragma unroll
  for (int d = 0; d < 8; ++d) e[d] = emb[n * DDIM + d];
#pragma unroll
  for (int d = 0; d < 8; ++d)
    nrm[d] = nan_to_num((e[d] - stats[8 + d]) / stats[d]);

  const float inv_sqrt_d = 0.35355339059327373f;
#pragma unroll
  for (int j = 0; j < 8; ++j) {
    float sq = 0.f, sk = 0.f, sv = 0.f;
#pragma unroll
    for (int d = 0; d < 8; ++d) {
      sq = fmaf(nrm[d], qw[j * 8 + d], sq);
      sk = fmaf(nrm[d], kw[j * 8 + d], sk);
      sv = fmaf(e[d],  vw[j * 8 + d], sv);
    }
    q[n * DDIM + j] = sq * inv_sqrt_d;   // scores pre-scaled
    k[n * DDIM + j] = sk;
    v[n * DDIM + j] = sv;
  }
}

// ---- flash attention + MLP + residual, one wave per 16-query tile ---------
__global__ __launch_bounds__(32) void k_attn(
    const float* __restrict__ q, const float* __restrict__ k,
    const float* __restrict__ v, float* __restrict__ emb,
    const float* __restrict__ pw, const float* __restrict__ pb,
    const float* __restrict__ ow, const float* __restrict__ ob) {
  const int lane  = threadIdx.x;
  const int col   = lane & 15;     // query index within tile
  const int hi    = lane >> 4;     // half-wave selector
  const int qbase = blockIdx.x * 16;
  const bool ishi = (hi != 0);

  // B operand (Q^T) for the score WMMA, loaded once:
  // chunk c -> float2 {Q[col][4c+2hi], Q[col][4c+2hi+1]}
  const float* qrow = q + (qbase + col) * DDIM + 2 * hi;
  v2f qb0 = *(const v2f*)(qrow + 0);
  v2f qb1 = *(const v2f*)(qrow + 4);

  float m = -3.4028235e38f;
  float lsum = 0.f;
  v8f acc = {};                    // O^T accumulator: VGPR r, lo lanes = O[q=lane][d=r]

  const float* vcol = v + (lane & 7);   // V column for this lane's d (rows 8-15 dup)

  for (int kt = 0; kt < NTOK / 16; ++kt) {
    const int kbase = kt * 16;

    // ---- S^T = K_tile x Q^T -------------------------------------------
    const float* krow = k + (kbase + col) * DDIM + 2 * hi;
    v2f ka0 = *(const v2f*)(krow + 0);
    v2f ka1 = *(const v2f*)(krow + 4);

    // prefetch next tile's K rows and V rows (global_prefetch_b8)
    __builtin_prefetch(krow + 16 * DDIM, 0, 1);
    __builtin_prefetch(vcol + (kbase + 16 + col) * DDIM, 0, 1);

    v8f S = {};
    S = __builtin_amdgcn_wmma_f32_16x16x4_f32(false, ka0, false, qb0,
                                              (short)0, S, false, false);
    S = __builtin_amdgcn_wmma_f32_16x16x4_f32(false, ka1, false, qb1,
                                              (short)0, S, false, false);
    // S layout: VGPR r, lane<16 -> [key=r][q=lane]; lane>=16 -> [key=r+8][q=lane-16]

    // ---- online softmax over keys -------------------------------------
    float tmax = S[0];
#pragma unroll
    for (int r = 1; r < 8; ++r) tmax = fmaxf(tmax, S[r]);
    tmax = fmaxf(tmax, __shfl_xor(tmax, 16, 32));
    float mnew  = fmaxf(m, tmax);
    float scale = __expf(m - mnew);

    float p[8];
    float tsum = 0.f;
#pragma unroll
    for (int r = 0; r < 8; ++r) {
      p[r] = __expf(S[r] - mnew);
      tsum += p[r];
    }
    tsum += __shfl_xor(tsum, 16, 32);
    lsum = lsum * scale + tsum;
    m = mnew;

#pragma unroll
    for (int r = 0; r < 8; ++r) acc[r] *= scale;   // rescale accumulator

    // ---- move P (C/D layout) into B-operand layout ---------------------
    // One shfl_xor(16) + one select per register.
    float ps[8];
#pragma unroll
    for (int r = 0; r < 8; ++r) ps[r] = __shfl_xor(p[r], 16, 32);

    v2f pB0 = { ishi ? ps[2] : p[0], ishi ? ps[3] : p[1] };   // keys 0-3
    v2f pB1 = { ishi ? ps[6] : p[4], ishi ? ps[7] : p[5] };   // keys 4-7
    v2f pB2 = { ishi ? p[2] : ps[0], ishi ? p[3] : ps[1] };   // keys 8-11
    v2f pB3 = { ishi ? p[6] : ps[4], ishi ? p[7] : ps[5] };   // keys 12-15

    // ---- A operand: V^T chunks (rows M=d; M>=8 harmless duplicates) ----
    v2f vA0 = { vcol[(kbase +  0 + 2 * hi) * DDIM], vcol[(kbase +  1 + 2 * hi) * DDIM] };
    v2f vA1 = { vcol[(kbase +  4 + 2 * hi) * DDIM], vcol[(kbase +  5 + 2 * hi) * DDIM] };
    v2f vA2 = { vcol[(kbase +  8 + 2 * hi) * DDIM], vcol[(kbase +  9 + 2 * hi) * DDIM] };
    v2f vA3 = { vcol[(kbase + 12 + 2 * hi) * DDIM], vcol[(kbase + 13 + 2 * hi) * DDIM] };

    // ---- O^T += V^T x P : 4 accumulating WMMAs -------------------------
    acc = __builtin_amdgcn_wmma_f32_16x16x4_f32(false, vA0, false, pB0,
                                                (short)0, acc, false, false);
    acc = __builtin_amdgcn_wmma_f32_16x16x4_f32(false, vA1, false, pB1,
                                                (short)0, acc, false, false);
    acc = __builtin_amdgcn_wmma_f32_16x16x4_f32(false, vA2, false, pB2,
                                                (short)0, acc, false, false);
    acc = __builtin_amdgcn_wmma_f32_16x16x4_f32(false, vA3, false, pB3,
                                                (short)0, acc, false, false);
  }

  // acc lo lanes already hold the full 16-key-summed O[q=lane][d=r]
  if (hi == 0) {
    const int n = qbase + col;
    const float rinv = 1.0f / lsum;
    float f[8], g[8], h[8];
#pragma unroll
    for (int d = 0; d < 8; ++d) f[d] = acc[d] * rinv;
#pragma unroll
    for (int j = 0; j < 8; ++j) {          // relu(f @ pre_w.T + pre_b)
      float s = pb[j];
#pragma unroll
      for (int d = 0; d < 8; ++d) s = fmaf(f[d], pw[j * 8 + d], s);
      g[j] = fmaxf(s, 0.0f);
    }
#pragma unroll
    for (int j = 0; j < 8; ++j) {          // g @ post_w.T + post_b
      float s = ob[j];
#pragma unroll
      for (int d = 0; d < 8; ++d) s = fmaf(g[d], ow[j * 8 + d], s);
      h[j] = s;
    }
#pragma unroll
    for (int d = 0; d < 8; ++d) {
      float e = emb[n * DDIM + d];
      emb[n * DDIM + d] = h[d] * 0.2f + e * 0.8f;
    }
  }
}

// ---- emb[:, :3] -> out [3,64,64] ------------------------------------------
__global__ __launch_bounds__(256) void k_out(const float* __restrict__ emb,
                                             float* __restrict__ out) {
  int i = blockIdx.x * 256 + threadIdx.x;   // 0 .. 12287
  if (i >= 3 * NTOK) return;
  int c  = i / NTOK;
  int ij = i - c * NTOK;
  out[i] = emb[ij * DDIM + c];
}

extern "C" void kernel_launch(void* const* d_in, const int* in_sizes, int n_in,
                              void* d_out, int out_size, void* d_ws,
                              size_t ws_size, hipStream_t stream) {
  const float* in     = (const float*)d_in[0];
  const float* qw     = (const float*)d_in[1];
  const float* kw     = (const float*)d_in[2];
  const float* vw     = (const float*)d_in[3];
  const float* pre_w  = (const float*)d_in[4];
  const float* pre_b  = (const float*)d_in[5];
  const float* post_w = (const float*)d_in[6];
  const float* post_b = (const float*)d_in[7];

  float* ws    = (float*)d_ws;
  float* emb   = ws;                 // 4096*8
  float* q     = ws + 32768;         // 4096*8
  float* k     = ws + 65536;         // 4096*8
  float* v     = ws + 98304;         // 4096*8
  float* stats = ws + 131072;        // 16 (mean[8], var[8])

  k_embed<<<128, 256, 0, stream>>>(in, emb);
  for (int l = 0; l < 8; ++l) {
    k_stats<<<1, 256, 0, stream>>>(emb, stats);
    k_qkv<<<16, 256, 0, stream>>>(emb, stats, qw + l * 64, kw + l * 64,
                                  vw + l * 64, q, k, v);
    k_attn<<<NTOK / 16, 32, 0, stream>>>(q, k, v, emb, pre_w + l * 64,
                                         pre_b + l * 8, post_w + l * 64,
                                         post_b + l * 8);
  }
  k_out<<<48, 256, 0, stream>>>(emb, (float*)d_out);
}